// MixtureOfDepthsRouter_17927193493872
// MI455X (gfx1250) — compile-verified
//
#include <hip/hip_runtime.h>
#include <hip/hip_bf16.h>
#include <math.h>

// ---------------------------------------------------------------------------
// MixtureOfDepths router for MI455X (gfx1250, wave32).
//
// Phase 1 (bandwidth-bound, 512 MB @ 23.3 TB/s ~ 22 us):
//   GEMV logits = hidden @ w via V_WMMA_F32_16X16X4_F32.
//   One wave handles 16 rows; lane L supplies A[M=L&15][K=k0+2*(L>>4)+{0,1}]
//   as a single b64 load (matches the ISA 16x4 f32 A layout exactly).
//   B[k][n] = w[k0+k] for all n  ->  every D column equals the dot product.
//   4 independent accumulator chains hide WMMA latency.
//
// Phase 2 (negligible, 128 KB): per-batch k-th-largest via 32-step binary
//   search on IEEE bit patterns (weights are sigmoid outputs > 0, so uint
//   bit order == value order), then mask = weights >= threshold.
// ---------------------------------------------------------------------------

typedef float v2f __attribute__((ext_vector_type(2)));
typedef float v8f __attribute__((ext_vector_type(8)));

#define HIDDEN     4096
#define SEQ        8192
#define BATCH      4
#define ROWS_TOTAL (BATCH * SEQ)   // 32768
#define KSEL       (SEQ / 2)       // 4096

__global__ __launch_bounds__(256) void mod_router_gemv_wmma(
    const float* __restrict__ hs,     // (4, 8192, 4096)
    const float* __restrict__ w,      // (4096,)
    const float* __restrict__ bptr,   // scalar
    float*       __restrict__ out_weights)   // (32768,)
{
    const int lane  = threadIdx.x & 31;
    const int wave  = (blockIdx.x * blockDim.x + threadIdx.x) >> 5;
    const int row_base = wave * 16;          // 16 rows per wave
    const int m     = lane & 15;             // M index this lane feeds
    const int khalf = lane >> 4;             // 0: K={0,1}, 1: K={2,3}

    const float* arow = hs + (size_t)(row_base + m) * HIDDEN + 2 * khalf;
    const float* wrow = w + 2 * khalf;

    v8f c0 = {}, c1 = {}, c2 = {}, c3 = {};

    for (int k0 = 0; k0 < HIDDEN; k0 += 16) {
        v2f a0 = *(const v2f*)(arow + k0);
        v2f a1 = *(const v2f*)(arow + k0 + 4);
        v2f a2 = *(const v2f*)(arow + k0 + 8);
        v2f a3 = *(const v2f*)(arow + k0 + 12);
        v2f b0 = *(const v2f*)(wrow + k0);
        v2f b1 = *(const v2f*)(wrow + k0 + 4);
        v2f b2 = *(const v2f*)(wrow + k0 + 8);
        v2f b3 = *(const v2f*)(wrow + k0 + 12);
        // 8 args: (neg_a, A, neg_b, B, c_mod, C, reuse_a, reuse_b)
        c0 = __builtin_amdgcn_wmma_f32_16x16x4_f32(false, a0, false, b0, (short)0, c0, false, false);
        c1 = __builtin_amdgcn_wmma_f32_16x16x4_f32(false, a1, false, b1, (short)0, c1, false, false);
        c2 = __builtin_amdgcn_wmma_f32_16x16x4_f32(false, a2, false, b2, (short)0, c2, false, false);
        c3 = __builtin_amdgcn_wmma_f32_16x16x4_f32(false, a3, false, b3, (short)0, c3, false, false);
    }

    v8f csum = c0 + c1 + c2 + c3;
    const float bias = bptr[0];

    // D layout: VGPR j holds M=j (lanes 0-15) / M=j+8 (lanes 16-31); all N
    // columns identical. Lane 0 writes rows 0..7, lane 16 writes rows 8..15.
    if (m == 0) {
        const int rbase = row_base + (khalf ? 8 : 0);
        #pragma unroll
        for (int j = 0; j < 8; ++j) {
            float logit = csum[j] + bias;
            float wgt   = 1.0f / (1.0f + expf(-logit));
            out_weights[rbase + j] = wgt;
        }
    }
}

__global__ __launch_bounds__(256) void mod_router_topk_mask(
    float* __restrict__ out)   // [0:32768] weights, [32768:65536] mask
{
    __shared__ unsigned sbits[SEQ];   // 32 KB of 320 KB LDS
    __shared__ unsigned scount;

    const int b = blockIdx.x;
    const float* wts = out + (size_t)b * SEQ;

    for (int i = threadIdx.x; i < SEQ; i += 256)
        sbits[i] = __float_as_uint(wts[i]);
    __syncthreads();

    // Binary search for max x with count(bits >= x) >= KSEL  == k-th largest.
    unsigned lo = 0u;            // count(>=lo) = SEQ >= KSEL
    unsigned hi = 0x7F800001u;   // above any finite positive float -> count < KSEL
    for (int iter = 0; iter < 32; ++iter) {
        unsigned mid = lo + ((hi - lo) >> 1);
        if (mid == lo) break;                 // uniform across block
        if (threadIdx.x == 0) scount = 0u;
        __syncthreads();
        unsigned cnt = 0;
        for (int i = threadIdx.x; i < SEQ; i += 256)
            cnt += (sbits[i] >= mid) ? 1u : 0u;
        atomicAdd(&scount, cnt);
        __syncthreads();
        unsigned total = scount;
        __syncthreads();
        if (total >= KSEL) lo = mid; else hi = mid;
    }

    const unsigned tbits = lo;
    float* mask = out + ROWS_TOTAL + (size_t)b * SEQ;
    for (int i = threadIdx.x; i < SEQ; i += 256)
        mask[i] = (sbits[i] >= tbits) ? 1.0f : 0.0f;
}

extern "C" void kernel_launch(void* const* d_in, const int* in_sizes, int n_in,
                              void* d_out, int out_size, void* d_ws, size_t ws_size,
                              hipStream_t stream) {
    const float* hs   = (const float*)d_in[0];   // (4,8192,4096) fp32
    const float* w    = (const float*)d_in[1];   // (4096,) fp32
    const float* bias = (const float*)d_in[2];   // scalar fp32
    float* out = (float*)d_out;

    // 32768 rows / 16 rows-per-wave / 8 waves-per-block = 256 blocks.
    mod_router_gemv_wmma<<<256, 256, 0, stream>>>(hs, w, bias, out);
    mod_router_topk_mask<<<BATCH, 256, 0, stream>>>(out);
}